// NllbMoeTop2Router_20572893348642
// MI455X (gfx1250) — compile-verified
//
#include <hip/hip_runtime.h>
#include <hip/hip_bf16.h>
#include <float.h>
#include <math.h>

#define TOKENS   16384
#define HIDDEN   2048
#define NEXP     64
#define CAPACITY 512   // 2 * ceil(16384/64)

typedef __attribute__((ext_vector_type(2))) float v2f;
typedef __attribute__((ext_vector_type(8))) float v8f;

// ---------------------------------------------------------------------------
// Kernel 1: router logits GEMM  (16384x2048) x (2048x64) -> (16384x64), f32.
// One wave computes a 16x16 tile with V_WMMA_F32_16X16X4_F32 over K=2048.
// Block = 256 threads = 8 waves = 32(M) x 64(N) tile. Grid = 512 blocks.
// ---------------------------------------------------------------------------
__global__ __launch_bounds__(256) void gemm_logits_kernel(
    const float* __restrict__ x, const float* __restrict__ w,
    float* __restrict__ logits) {
  const int tid   = threadIdx.x;
  const int lane  = tid & 31;
  const int wv    = tid >> 5;       // 0..7
  const int mtile = wv >> 2;        // 0..1
  const int ntile = wv & 3;         // 0..3
  const int l     = lane & 15;      // lane within half-wave
  const int h     = lane >> 4;      // 0 or 1
  const int rowBase = blockIdx.x * 32 + mtile * 16;
  const int colBase = ntile * 16;

  // A layout (16x4 f32): lane l(+16h) holds A[M=l][K=2h], A[M=l][K=2h+1]
  const float* aRow = x + (size_t)(rowBase + l) * HIDDEN + 2 * h;
  // B layout (4x16 f32): lane l(+16h) holds B[K=2h][N=l], B[K=2h+1][N=l]
  const float* bCol = w + (size_t)(2 * h) * NEXP + colBase + l;

  v8f acc = {};
  for (int k0 = 0; k0 < HIDDEN; k0 += 32) {
    // stream-ahead prefetch on the A row (speculative; drops on miss)
    __builtin_prefetch(aRow + ((k0 + 512) & (HIDDEN - 1)), 0, 3);
#pragma unroll
    for (int kk = 0; kk < 32; kk += 4) {
      const int k = k0 + kk;
      v2f a = *(const v2f*)(aRow + k);        // global_load_b64
      v2f b;
      b.x = bCol[(size_t)k * NEXP];           // W[k+2h     ][col]
      b.y = bCol[(size_t)(k + 1) * NEXP];     // W[k+2h + 1 ][col]
      acc = __builtin_amdgcn_wmma_f32_16x16x4_f32(
          false, a, false, b, (short)0, acc, false, false);
    }
  }

  // D layout: VGPR r -> row (r + 8h), col l
  float* outP = logits + (size_t)(rowBase + 8 * h) * NEXP + colBase + l;
#pragma unroll
  for (int r = 0; r < 8; ++r) outP[(size_t)r * NEXP] = acc[r];
}

// ---------------------------------------------------------------------------
// Kernel 2: wave-per-token top-2 + softmax. Lane handles experts lane, lane+32.
// Emits p1, p2, packed (e1|e2<<8).
// ---------------------------------------------------------------------------
__global__ __launch_bounds__(256) void router_top2_kernel(
    const float* __restrict__ logits,
    float* __restrict__ p1o, float* __restrict__ p2o,
    unsigned int* __restrict__ e12o) {
  const int lane  = threadIdx.x & 31;
  const int token = blockIdx.x * 8 + (threadIdx.x >> 5);
  const float* row = logits + (size_t)token * NEXP;
  const float l0 = row[lane];
  const float l1 = row[lane + 32];

  // argmax (first-occurrence tie-break, like jnp.argmax)
  float v = l0; int idx = lane;
  if (l1 > v) { v = l1; idx = lane + 32; }
#pragma unroll
  for (int off = 16; off >= 1; off >>= 1) {
    float ov = __shfl_xor(v, off, 32);
    int   oi = __shfl_xor(idx, off, 32);
    if (ov > v || (ov == v && oi < idx)) { v = ov; idx = oi; }
  }
  const int e1 = idx; const float m = v;

  // second argmax with top-1 masked to -inf
  float m0 = (lane      == e1) ? -__builtin_inff() : l0;
  float m1 = (lane + 32 == e1) ? -__builtin_inff() : l1;
  v = m0; idx = lane;
  if (m1 > v) { v = m1; idx = lane + 32; }
#pragma unroll
  for (int off = 16; off >= 1; off >>= 1) {
    float ov = __shfl_xor(v, off, 32);
    int   oi = __shfl_xor(idx, off, 32);
    if (ov > v || (ov == v && oi < idx)) { v = ov; idx = oi; }
  }
  const int e2 = idx; const float l2 = v;

  // softmax denominator (shifted by max)
  float s = __expf(l0 - m) + __expf(l1 - m);
#pragma unroll
  for (int off = 16; off >= 1; off >>= 1) s += __shfl_xor(s, off, 32);

  if (lane == 0) {
    p1o[token] = 1.0f / s;            // exp(m - m)/s
    p2o[token] = __expf(l2 - m) / s;
    e12o[token] = (unsigned)e1 | ((unsigned)e2 << 8);
  }
}

// ---------------------------------------------------------------------------
// Kernel 3: one workgroup per expert. Build this expert's token list in a
// 128 KB dynamic-LDS key array (key = ~bits(p1) : token, ascending = routing
// order), rank by counting smaller keys, apply capacity. The top-1 list
// length is exactly the locations2 offset for this expert.
// ---------------------------------------------------------------------------
__global__ __launch_bounds__(256) void rank_capacity_kernel(
    const float* __restrict__ p1, const unsigned int* __restrict__ e12,
    unsigned int* __restrict__ keep1, unsigned int* __restrict__ keep2) {
  extern __shared__ unsigned long long key[];   // TOKENS entries worst case
  __shared__ int cnt;
  __shared__ int cnt1s;
  const unsigned e = blockIdx.x;
  const int tid = threadIdx.x;

  if (tid == 0) cnt = 0;
  __syncthreads();
  // phase A: collect top-1 tokens of this expert
  for (int t = tid; t < TOKENS; t += 256) {
    if ((e12[t] & 0xFFu) == e) {
      int pos = atomicAdd(&cnt, 1);
      key[pos] = ((unsigned long long)(0xFFFFFFFFu - __float_as_uint(p1[t])) << 32)
                 | (unsigned)t;
    }
  }
  __syncthreads();
  const int n1 = cnt;
  if (tid == 0) cnt1s = n1;
  // phase B: rank + capacity for top-1
  for (int i = tid; i < n1; i += 256) {
    const unsigned long long ki = key[i];
    int r = 0;
    for (int j = 0; j < n1; ++j) r += (key[j] < ki) ? 1 : 0;
    keep1[(unsigned)ki] = (r < CAPACITY) ? 1u : 0u;
  }
  __syncthreads();
  if (tid == 0) cnt = 0;
  __syncthreads();
  // phase C: collect top-2 tokens of this expert
  for (int t = tid; t < TOKENS; t += 256) {
    if (((e12[t] >> 8) & 0xFFu) == e) {
      int pos = atomicAdd(&cnt, 1);
      key[pos] = ((unsigned long long)(0xFFFFFFFFu - __float_as_uint(p1[t])) << 32)
                 | (unsigned)t;
    }
  }
  __syncthreads();
  const int n2 = cnt;
  const int base = cnt1s;   // sum of (pre-capacity) top-1 mask for this expert
  for (int i = tid; i < n2; i += 256) {
    const unsigned long long ki = key[i];
    int r = 0;
    for (int j = 0; j < n2; ++j) r += (key[j] < ki) ? 1 : 0;
    keep2[(unsigned)ki] = ((r + base) < CAPACITY) ? 1u : 0u;
  }
}

// ---------------------------------------------------------------------------
// Kernel 4: write top_1_mask and combined gates (coalesced, 64 thr / token).
// ---------------------------------------------------------------------------
__global__ __launch_bounds__(256) void write_out_kernel(
    const float* __restrict__ p1, const float* __restrict__ p2,
    const unsigned int* __restrict__ e12,
    const unsigned int* __restrict__ keep1, const unsigned int* __restrict__ keep2,
    float* __restrict__ out) {
  const int tid   = threadIdx.x;
  const int token = blockIdx.x * 4 + (tid >> 6);
  const int j     = tid & 63;
  const unsigned pack = e12[token];
  const int e1 = pack & 0xFF;
  const int e2 = (pack >> 8) & 0xFF;
  const unsigned k1 = keep1[token];
  const unsigned k2 = keep2[token];
  const float t1 = k1 ? p1[token] : 0.0f;
  const float t2 = k2 ? p2[token] : 0.0f;
  const float denom = fmaxf(t1 + t2, FLT_EPSILON);  // clip after dropping
  const float g1 = t1 / denom;
  const float g2 = t2 / denom;

  out[(size_t)token * NEXP + j] = (j == e1 && k1) ? 1.0f : 0.0f;
  out[(size_t)TOKENS * NEXP + (size_t)token * NEXP + j] =
      (j == e1) ? g1 : ((j == e2) ? g2 : 0.0f);
}

// ---------------------------------------------------------------------------
extern "C" void kernel_launch(void* const* d_in, const int* in_sizes, int n_in,
                              void* d_out, int out_size, void* d_ws, size_t ws_size,
                              hipStream_t stream) {
  const float* x = (const float*)d_in[0];  // (8,2048,2048) f32
  const float* w = (const float*)d_in[1];  // (2048,64) f32

  char* ws = (char*)d_ws;
  float* logits = (float*)ws;                                   // 4 MB
  float* p1 = (float*)(ws + (size_t)TOKENS * NEXP * sizeof(float));
  float* p2 = p1 + TOKENS;
  unsigned int* e12   = (unsigned int*)(p2 + TOKENS);
  unsigned int* keep1 = e12 + TOKENS;
  unsigned int* keep2 = keep1 + TOKENS;

  gemm_logits_kernel<<<TOKENS / 32, 256, 0, stream>>>(x, w, logits);
  router_top2_kernel<<<TOKENS / 8, 256, 0, stream>>>(logits, p1, p2, e12);
  rank_capacity_kernel<<<NEXP, 256, (size_t)TOKENS * 8, stream>>>(p1, e12, keep1, keep2);
  write_out_kernel<<<TOKENS / 4, 256, 0, stream>>>(p1, p2, e12, keep1, keep2,
                                                   (float*)d_out);
}